// TarsCoreBlock_44263932953093
// MI455X (gfx1250) — compile-verified
//
#include <hip/hip_runtime.h>
#include <hip/hip_bf16.h>

typedef float v2f __attribute__((ext_vector_type(2)));
typedef float v8f __attribute__((ext_vector_type(8)));

// D(16x16,f32) = A(16x4,f32) x B(4x16,f32) + C ; full-precision CDNA5 WMMA.
__device__ __forceinline__ v8f wmma4(float ax, float ay, float bx, float by, v8f acc) {
  v2f a; a.x = ax; a.y = ay;
  v2f b; b.x = bx; b.y = by;
  return __builtin_amdgcn_wmma_f32_16x16x4_f32(false, a, false, b, (short)0, acc,
                                               false, false);
}

#define ZERO8 {0.f,0.f,0.f,0.f,0.f,0.f,0.f,0.f}

// ---------------------------------------------------------------------------
// RWKV-7 WKV scan: 32 sequences, state row i owned by thread i (64 VGPRs).
// state[i,j] = state[i,j]*w[i] + (b[i]*k[i])*delta[j];  y[i]=r[i]*sum_j s[i,j]k[j]
// ---------------------------------------------------------------------------
__global__ __launch_bounds__(64) void wkv_kernel(
    const float* __restrict__ r, const float* __restrict__ k,
    const float* __restrict__ v, const float* __restrict__ w,
    const float* __restrict__ bonus, float* __restrict__ out)
{
  const int batch = blockIdx.x;          // 0..31  (= b*16 + h)
  const int i = threadIdx.x;             // 0..63  state row
  const int bb = batch >> 4;
  const int hh = batch & 15;

  __shared__ float kbuf[64];
  __shared__ float dbuf[64];

  float s[64];
#pragma unroll
  for (int j = 0; j < 64; ++j) s[j] = 0.f;

  const size_t ib = (size_t)batch * 2048u * 64u + (size_t)i;
  const size_t ob = (size_t)bb * 2048u * 1024u + (size_t)hh * 64u + (size_t)i;

  for (int t = 0; t < 2048; ++t) {
    const size_t o = ib + (size_t)t * 64u;
    const float kt = k[o];
    const float vt = v[o];
    const float wt = w[o];
    const float bt = bonus[o];
    const float rt = r[o];
    if (t < 2047) {                      // wave-uniform; hide next step's latency
      __builtin_prefetch(k + o + 64, 0, 1);
      __builtin_prefetch(v + o + 64, 0, 1);
      __builtin_prefetch(w + o + 64, 0, 1);
      __builtin_prefetch(bonus + o + 64, 0, 1);
      __builtin_prefetch(r + o + 64, 0, 1);
    }
    __syncthreads();                     // WAR: prev iter finished reading kbuf/dbuf
    kbuf[i] = kt;
    __syncthreads();
    float Sk = 0.f;
#pragma unroll
    for (int j = 0; j < 64; ++j) Sk += s[j] * kbuf[j];
    dbuf[i] = vt - Sk;
    __syncthreads();
    const float bk = bt * kt;
    float y = 0.f;
#pragma unroll
    for (int j = 0; j < 64; ++j) {
      s[j] = s[j] * wt + bk * dbuf[j];
      y += s[j] * kbuf[j];
    }
    out[ob + (size_t)t * 1024u] = rt * y;   // full coverage of d_out ('=' init)
  }
}

// ---------------------------------------------------------------------------
// SSD pass 1: per (b,h,chunk): Y_diag += out, chunk_states -> ws
// ---------------------------------------------------------------------------
__global__ __launch_bounds__(256) void ssd_chunk_kernel(
    const float* __restrict__ X, const float* __restrict__ A,
    const float* __restrict__ Bm, const float* __restrict__ Cm,
    float* __restrict__ out, float* __restrict__ ws)
{
  const int gid = blockIdx.x;            // 1024 = 2*16*32
  const int c  = gid & 31;
  const int h  = (gid >> 5) & 15;
  const int bb = gid >> 9;

  __shared__ float BsGs[64][65];         // B chunk; reused for masked G (A-operand)
  __shared__ float Cs[64][65];
  __shared__ float Xs[64][64];           // only row-contiguous reads -> no pad
  __shared__ float sAc[64];

  const int tid  = threadIdx.x;
  const int wv   = tid >> 5;
  const int lane = tid & 31;
  const int lo   = lane & 15;
  const int hi   = lane >> 4;
  const int koff = hi << 1;

  const size_t chunkBase = ((size_t)bb * 2048u + (size_t)c * 64u) * 16u + (size_t)h;
  const float* Xg = X  + chunkBase * 64u;
  const float* Bg = Bm + chunkBase * 64u;
  const float* Cg = Cm + chunkBase * 64u;
  const float* Ag = A  + chunkBase;

  for (int e = tid; e < 1024; e += 256) {
    const int row = e >> 4;
    const int c4  = (e & 15) << 2;
    const size_t go = (size_t)row * 1024u + (size_t)c4;
    const float4 bx = *(const float4*)(Bg + go);
    const float4 cx = *(const float4*)(Cg + go);
    const float4 xx = *(const float4*)(Xg + go);
    BsGs[row][c4+0]=bx.x; BsGs[row][c4+1]=bx.y; BsGs[row][c4+2]=bx.z; BsGs[row][c4+3]=bx.w;
    Cs[row][c4+0]=cx.x;   Cs[row][c4+1]=cx.y;   Cs[row][c4+2]=cx.z;   Cs[row][c4+3]=cx.w;
    Xs[row][c4+0]=xx.x;   Xs[row][c4+1]=xx.y;   Xs[row][c4+2]=xx.z;   Xs[row][c4+3]=xx.w;
  }
  if (tid < 64) sAc[tid] = Ag[(size_t)tid * 16u];
  __syncthreads();
  if (tid == 0) {                        // inclusive cumsum of A over the chunk
    float run = 0.f;
    for (int t = 0; t < 64; ++t) { run += sAc[t]; sAc[t] = run; }
  }
  __syncthreads();

  // Phase 2 (regs): per wave, 2 tiles of G = C*B^T and 2 tiles of chunk-states
  v8f g[2];
  v8f st[2];
  const float aLast = sAc[63];
#pragma unroll
  for (int q = 0; q < 2; ++q) {
    const int tIdx = wv * 2 + q;
    const int tm = tIdx >> 2;
    const int tn = tIdx & 3;

    v8f acc = ZERO8;                     // G[l,s] = sum_n C[l,n]*B[s,n]
#pragma unroll
    for (int kk = 0; kk < 64; kk += 4) {
      const int k0 = kk + koff;
      acc = wmma4(Cs[tm*16+lo][k0],   Cs[tm*16+lo][k0+1],
                  BsGs[tn*16+lo][k0], BsGs[tn*16+lo][k0+1], acc);
    }
    g[q] = acc;

    v8f acc2 = ZERO8;                    // states[p,n] = sum_l X[l,p]*decay[l]*B[l,n]
#pragma unroll
    for (int kk = 0; kk < 64; kk += 4) {
      const int k0 = kk + koff;
      const float d0 = __expf(aLast - sAc[k0]);
      const float d1 = __expf(aLast - sAc[k0+1]);
      acc2 = wmma4(Xs[k0][tm*16+lo] * d0, Xs[k0+1][tm*16+lo] * d1,
                   BsGs[k0][tn*16+lo],    BsGs[k0+1][tn*16+lo], acc2);
    }
    st[q] = acc2;
  }

  // chunk states -> workspace [bh][c][p][n]
  float* wsb = ws + ((size_t)(bb * 16 + h) * 32u + (size_t)c) * 4096u;
#pragma unroll
  for (int q = 0; q < 2; ++q) {
    const int tIdx = wv * 2 + q, tm = tIdx >> 2, tn = tIdx & 3;
#pragma unroll
    for (int rr = 0; rr < 8; ++rr) {
      const int p = tm * 16 + rr + hi * 8;
      const int n = tn * 16 + lo;
      wsb[p * 64 + n] = st[q][rr];
    }
  }

  __syncthreads();                       // all reads of B done; retarget buffer to G
#pragma unroll
  for (int q = 0; q < 2; ++q) {          // masked G (apply exp segsum / causal mask)
    const int tIdx = wv * 2 + q, tm = tIdx >> 2, tn = tIdx & 3;
#pragma unroll
    for (int rr = 0; rr < 8; ++rr) {
      const int l = tm * 16 + rr + hi * 8;
      const int s = tn * 16 + lo;
      const float f = (s <= l) ? __expf(sAc[l] - sAc[s]) : 0.f;
      BsGs[l][s] = g[q][rr] * f;
    }
  }
  __syncthreads();

  // Y_diag = G @ X, accumulate into out (wkv already wrote '=')
  float* og = out + ((size_t)bb * 2048u + (size_t)c * 64u) * 1024u + (size_t)h * 64u;
#pragma unroll
  for (int q = 0; q < 2; ++q) {
    const int tIdx = wv * 2 + q, tm = tIdx >> 2, tn = tIdx & 3;
    v8f acc = ZERO8;
#pragma unroll
    for (int kk = 0; kk < 64; kk += 4) {
      const int k0 = kk + koff;
      acc = wmma4(BsGs[tm*16+lo][k0], BsGs[tm*16+lo][k0+1],
                  Xs[k0][tn*16+lo],   Xs[k0+1][tn*16+lo], acc);
    }
#pragma unroll
    for (int rr = 0; rr < 8; ++rr) {
      const int l = tm * 16 + rr + hi * 8;
      const int p = tn * 16 + lo;
      const size_t off = (size_t)l * 1024u + (size_t)p;
      og[off] += acc[rr];
    }
  }
}

// ---------------------------------------------------------------------------
// SSD pass 2: inter-chunk state scan (in place in ws): ws[c] <- state BEFORE chunk c
// ---------------------------------------------------------------------------
__global__ __launch_bounds__(256) void ssd_scan_kernel(
    const float* __restrict__ A, float* __restrict__ ws)
{
  const int bh = blockIdx.x;             // 0..31
  const int bb = bh >> 4;
  const int h  = bh & 15;
  const int tid = threadIdx.x;

  __shared__ float sdec[32];
  if (tid < 32) {
    float sum = 0.f;
    const float* Ag = A + ((size_t)bb * 2048u + (size_t)tid * 64u) * 16u + (size_t)h;
#pragma unroll 8
    for (int t = 0; t < 64; ++t) sum += Ag[(size_t)t * 16u];
    sdec[tid] = __expf(sum);             // exp(total A of chunk tid)
  }
  __syncthreads();

  float carry[16];
#pragma unroll
  for (int j = 0; j < 16; ++j) carry[j] = 0.f;

  float* base = ws + (size_t)bh * 32u * 4096u;
  for (int c = 0; c < 32; ++c) {
    const float d = sdec[c];
    float* p = base + (size_t)c * 4096u;
#pragma unroll
    for (int j = 0; j < 16; ++j) {
      const int idx = tid + (j << 8);
      const float tmp = p[idx];
      p[idx] = carry[j];                 // states_prev for chunk c
      carry[j] = carry[j] * d + tmp;
    }
  }
}

// ---------------------------------------------------------------------------
// SSD pass 3: Y_off[l,p] = exp(Acum[l]) * sum_n C[l,n] * states_prev[p,n]
// ---------------------------------------------------------------------------
__global__ __launch_bounds__(256) void ssd_yoff_kernel(
    const float* __restrict__ Cm, const float* __restrict__ A,
    const float* __restrict__ ws, float* __restrict__ out)
{
  const int gid = blockIdx.x;
  const int c  = gid & 31;
  const int h  = (gid >> 5) & 15;
  const int bb = gid >> 9;

  __shared__ float Cs[64][65];
  __shared__ float Sps[64][65];          // states_prev [p][n]
  __shared__ float sAc[64];

  const int tid  = threadIdx.x;
  const int wv   = tid >> 5;
  const int lane = tid & 31;
  const int lo   = lane & 15;
  const int hi   = lane >> 4;
  const int koff = hi << 1;

  const size_t chunkBase = ((size_t)bb * 2048u + (size_t)c * 64u) * 16u + (size_t)h;
  const float* Cg = Cm + chunkBase * 64u;
  const float* Ag = A  + chunkBase;
  const float* wsb = ws + ((size_t)(bb * 16 + h) * 32u + (size_t)c) * 4096u;

  for (int e = tid; e < 1024; e += 256) {
    const int row = e >> 4;
    const int c4  = (e & 15) << 2;
    const float4 cx = *(const float4*)(Cg + (size_t)row * 1024u + c4);
    const float4 sx = *(const float4*)(wsb + (size_t)row * 64u + c4);
    Cs[row][c4+0]=cx.x;  Cs[row][c4+1]=cx.y;  Cs[row][c4+2]=cx.z;  Cs[row][c4+3]=cx.w;
    Sps[row][c4+0]=sx.x; Sps[row][c4+1]=sx.y; Sps[row][c4+2]=sx.z; Sps[row][c4+3]=sx.w;
  }
  if (tid < 64) sAc[tid] = Ag[(size_t)tid * 16u];
  __syncthreads();
  if (tid == 0) {
    float run = 0.f;
    for (int t = 0; t < 64; ++t) { run += sAc[t]; sAc[t] = run; }
  }
  __syncthreads();

  float* og = out + ((size_t)bb * 2048u + (size_t)c * 64u) * 1024u + (size_t)h * 64u;
#pragma unroll
  for (int q = 0; q < 2; ++q) {
    const int tIdx = wv * 2 + q, tm = tIdx >> 2, tn = tIdx & 3;
    const float el = __expf(sAc[tm * 16 + lo]);   // row scale folded into A operand
    v8f acc = ZERO8;
#pragma unroll
    for (int kk = 0; kk < 64; kk += 4) {
      const int k0 = kk + koff;
      acc = wmma4(Cs[tm*16+lo][k0] * el, Cs[tm*16+lo][k0+1] * el,
                  Sps[tn*16+lo][k0],     Sps[tn*16+lo][k0+1], acc);
    }
#pragma unroll
    for (int rr = 0; rr < 8; ++rr) {
      const int l = tm * 16 + rr + hi * 8;
      const int p = tn * 16 + lo;
      const size_t off = (size_t)l * 1024u + (size_t)p;
      og[off] += acc[rr];
    }
  }
}

// ---------------------------------------------------------------------------
extern "C" void kernel_launch(void* const* d_in, const int* in_sizes, int n_in,
                              void* d_out, int out_size, void* d_ws, size_t ws_size,
                              hipStream_t stream) {
  (void)in_sizes; (void)n_in; (void)out_size; (void)ws_size;
  const float* X  = (const float*)d_in[0];
  const float* A  = (const float*)d_in[1];
  const float* Bm = (const float*)d_in[2];
  const float* Cm = (const float*)d_in[3];
  const float* r  = (const float*)d_in[4];
  const float* k  = (const float*)d_in[5];
  const float* v  = (const float*)d_in[6];
  const float* w  = (const float*)d_in[7];
  const float* bo = (const float*)d_in[8];
  float* out = (float*)d_out;
  float* ws  = (float*)d_ws;             // needs 32*32*4096*4 = 16 MiB scratch

  wkv_kernel      <<<32,   64, 0, stream>>>(r, k, v, w, bo, out);   // writes '='
  ssd_chunk_kernel<<<1024, 256, 0, stream>>>(X, A, Bm, Cm, out, ws);
  ssd_scan_kernel <<<32,   256, 0, stream>>>(A, ws);
  ssd_yoff_kernel <<<1024, 256, 0, stream>>>(Cm, A, ws, out);
}